// KernelConv2D_10797547782740
// MI455X (gfx1250) — compile-verified
//
#include <hip/hip_runtime.h>
#include <stdint.h>

// Problem constants (from reference)
constexpr int kB = 4;
constexpr int kC = 32;
constexpr int kH = 128;
constexpr int kW = 128;
constexpr int kK = 5;
constexpr int kPad = 2;

// Tiling: one block handles one (b,c) plane x ROWS output rows.
constexpr int ROWS = 8;                 // output rows per block
constexpr int TILE_R = ROWS + 2 * kPad; // 12 LDS rows (with halo)
constexpr int TILE_C = kW + 2 * kPad;   // 132 valid LDS cols
constexpr int TILE_STRIDE = 136;        // 136*4B = 544B row pitch -> 16B aligned
constexpr int BLOCK = 256;              // 8 wave32 waves

typedef float v4f __attribute__((ext_vector_type(4)));

// Raw LDS byte offset (addrspacecast generic -> LDS, then ptr->int).
__device__ __forceinline__ unsigned lds_byte_offset(const void* p) {
  return (unsigned)(size_t)(__attribute__((address_space(3))) const void*)p;
}

__global__ __launch_bounds__(BLOCK) void dynconv_kernel(
    const float* __restrict__ inp,   // [B, C, H, W]
    const float* __restrict__ kern,  // [B, C*25, H, W]
    float* __restrict__ out) {       // [B, C, H, W]
  // 16B-aligned so window reads are guaranteed ds_load_b128.
  __shared__ alignas(16) float tile[TILE_R * TILE_STRIDE];

  constexpr int tiles_per_bc = kH / ROWS;          // 16
  const int bc  = blockIdx.x / tiles_per_bc;       // 0 .. B*C-1
  const int h0  = (blockIdx.x % tiles_per_bc) * ROWS;
  const int tid = threadIdx.x;

  const float* src = inp + (size_t)bc * kH * kW;

  // ---- Stage input tile (with replicated/edge halo) into LDS via the
  //      CDNA5 async global->LDS path (ASYNCcnt). Per-lane clamped addresses
  //      implement ReplicationPad2d once per tile. ----
  constexpr int n_elems = TILE_R * TILE_C;         // 1584
#pragma unroll
  for (int base = 0; base < n_elems; base += BLOCK) {
    int idx = base + tid;
    if (idx >= n_elems) idx = n_elems - 1;         // benign duplicate load
    const int rt = idx / TILE_C;
    const int ct = idx - rt * TILE_C;
    int gr = h0 - kPad + rt;
    gr = gr < 0 ? 0 : (gr > kH - 1 ? kH - 1 : gr);
    int gc = ct - kPad;
    gc = gc < 0 ? 0 : (gc > kW - 1 ? kW - 1 : gc);
    const unsigned goff = (unsigned)((gr * kW + gc) * 4);
    const unsigned loff = lds_byte_offset(&tile[rt * TILE_STRIDE + ct]);
    asm volatile("global_load_async_to_lds_b32 %0, %1, %2"
                 :
                 : "v"(loff), "v"(goff), "s"(src)
                 : "memory");
  }
  asm volatile("s_wait_asynccnt 0" ::: "memory");
  __syncthreads();

  // ---- Compute: each thread -> float4 of one output row ----
  const int r  = tid >> 5;         // 0..7   (row within tile)
  const int w4 = (tid & 31) << 2;  // 0..124 (col, multiple of 4)
  const int h  = h0 + r;

  // kernel base: ((25*bc + t)*H + h)*W + w4, tap t adds t*H*W
  const float* kp = kern + ((size_t)25 * bc * kH + h) * kW + w4;
  constexpr size_t plane = (size_t)kH * kW;

  v4f acc = {0.f, 0.f, 0.f, 0.f};
#pragma unroll
  for (int i = 0; i < kK; ++i) {
    // 8-float sliding window: LDS cols w4 .. w4+7 of tile row r+i.
    // Row pitch 544B and w4*4B are 16B multiples -> two ds_load_b128.
    const v4f* trow = (const v4f*)&tile[(r + i) * TILE_STRIDE + w4];
    const v4f wlo = trow[0];
    const v4f whi = trow[1];
    const float win[8] = {wlo.x, wlo.y, wlo.z, wlo.w,
                          whi.x, whi.y, whi.z, whi.w};

#pragma unroll
    for (int j = 0; j < kK; ++j) {
      // Streamed (non-temporal) 16B/lane load of per-pixel weights:
      // zero reuse -> keep the 200MB kernel tensor out of L2 so the 8MB
      // input plane stays resident across all 25 taps.
      const v4f kv =
          __builtin_nontemporal_load((const v4f*)(kp + (size_t)(i * kK + j) * plane));
      acc.x += kv.x * win[j + 0];
      acc.y += kv.y * win[j + 1];
      acc.z += kv.z * win[j + 2];
      acc.w += kv.w * win[j + 3];
    }
  }

  float* op = out + ((size_t)bc * kH + h) * kW + w4;
  __builtin_nontemporal_store(acc, (v4f*)op);
}

extern "C" void kernel_launch(void* const* d_in, const int* in_sizes, int n_in,
                              void* d_out, int out_size, void* d_ws, size_t ws_size,
                              hipStream_t stream) {
  const float* inp  = (const float*)d_in[0];
  const float* kern = (const float*)d_in[1];
  float* out        = (float*)d_out;

  const int grid = kB * kC * (kH / ROWS);  // 2048 blocks
  dynconv_kernel<<<grid, BLOCK, 0, stream>>>(inp, kern, out);
}